// MyNet_66907000537542
// MI455X (gfx1250) — compile-verified
//
#include <hip/hip_runtime.h>

// CDNA5 / gfx1250: wave32, WMMA. Full-precision fp32 matrix op:
//   V_WMMA_F32_16X16X4_F32 : D(16x16 f32) = A(16x4 f32) x B(4x16 f32) + C
typedef __attribute__((ext_vector_type(2))) float v2f;
typedef __attribute__((ext_vector_type(8))) float v8f;
typedef int v4i_vec __attribute__((vector_size(16)));   // async-LDS builtin param type

static constexpr int HH    = 224;
static constexpr int WWID  = 224;
static constexpr int NB    = 16;
static constexpr int WTILE = 32;         // pixels along W per block
static constexpr int PATCH = WTILE + 2;  // 34 (halo)
static constexpr int ROWS  = 3;
static constexpr int RC    = ROWS * PATCH;  // 102

#if __has_builtin(__builtin_amdgcn_global_load_async_to_lds_b128) && \
    __has_builtin(__builtin_amdgcn_s_wait_asynccnt)
#define USE_ASYNC_LDS 1
#else
#define USE_ASYNC_LDS 0
#endif

// One-time weight re-layout (~1.2 MB total, L2-resident):
//   w[co][ci][kh][kw] -> wt[kh*3+kw][ ((ci>>1)*COUT + co)*2 + (ci&1) ]
// Each tap slice (and each ci-half of it) is a contiguous block in the exact
// pair-interleaved order the WMMA A-fragment wants in LDS.
template<int CIN, int COUT>
__global__ __launch_bounds__(256)
void wtransform(const float* __restrict__ w, float* __restrict__ wt) {
    const int i = blockIdx.x * 256 + threadIdx.x;
    constexpr int TOTAL = COUT * CIN * 9;
    if (i >= TOTAL) return;
    const int kw = i % 3;
    const int kh = (i / 3) % 3;
    const int ci = (i / 9) % CIN;
    const int co = i / (9 * CIN);
    wt[(kh * 3 + kw) * (CIN * COUT) + ((ci >> 1) * COUT + co) * 2 + (ci & 1)] = w[i];
}

// Implicit-GEMM 3x3 conv (stride 1, pad 1); one (n, h, 32-wide strip) per block.
// Per 16x16 WMMA tile: M = out-channels (A = weights), N = pixels (B = input).
// Weights stream through LDS in 18 double-buffered async phases (ASYNCcnt
// pipelined, 2 phases in flight); K-loop fully unrolled -> immediate DS offsets.
template<int CIN, int COUT>
__global__ __launch_bounds__(256)
void conv3x3_wmma(const float* __restrict__ x,
                  const float* __restrict__ wt,
                  float* __restrict__ out)
{
    constexpr int TPW    = (COUT / 16) * (WTILE / 16) / 8;  // co-tiles per wave: 2 or 1
    constexpr int HALF   = (CIN / 2) * COUT;                // floats per phase (ci-half of a tap)
    constexpr int PHOPS  = HALF / 4 / 256;                  // async b128 issues per thread per phase
    constexpr int NPH    = 18;                              // 9 taps x 2 ci-halves
    constexpr int KSTEPS = CIN / 8;                         // K=4 steps per phase

    __shared__ alignas(16) float sX[CIN * RC];    // ((ci>>1)*RC + rc)*2 + (ci&1)
    __shared__ alignas(16) float sW[2][HALF];     // double buffer; buf index == ci-half

    const int tid   = threadIdx.x;
    const int lane  = tid & 31;
    const int lan16 = lane & 15;
    const int hi16  = lane >> 4;     // K-half (ci pair) this lane holds
    const int wv    = tid >> 5;      // 0..7
    const int wSub  = wv & 1;        // 16-pixel subtile
    const int coT0  = wv >> 1;       // base co tile; second tile = coT0 + 4

    const int n  = blockIdx.z;
    const int h  = blockIdx.y;
    const int wB = blockIdx.x * WTILE;

    v8f acc[TPW];
#pragma unroll
    for (int t = 0; t < TPW; ++t) { v8f z = {}; acc[t] = z; }

    // ---- async weight staging: phase p = (tap = p>>1, ci-half = p&1)
    auto issueW = [&](int p) {
#if USE_ASYNC_LDS
        v4i_vec* g = (v4i_vec*)(wt + (size_t)p * HALF);
        v4i_vec* l = (v4i_vec*)sW[p & 1];
#pragma unroll
        for (int i = 0; i < PHOPS; ++i)
            __builtin_amdgcn_global_load_async_to_lds_b128(
                g + tid + i * 256, l + tid + i * 256, 0, 0);
#endif
    };
#if !USE_ASYNC_LDS
    auto copyW = [&](int p) {
        const float4* s = (const float4*)(wt + (size_t)p * HALF);
        float4*       d = (float4*)sW[p & 1];
        for (int i = tid; i < HALF / 4; i += 256) d[i] = s[i];
    };
#endif

    // ---- fully-unrolled compute for one phase: all DS offsets are immediates
    auto compute = [&](int p) {
        const int tap = p >> 1, half = p & 1;
        const int kh = tap / 3, kw = tap % 3;
        const v2f* __restrict__ Bp = (const v2f*)sX
            + (hi16 + half * (CIN / 4)) * RC + kh * PATCH + kw + wSub * 16 + lan16;
        const v2f* __restrict__ Ap = (const v2f*)sW[half]
            + hi16 * COUT + coT0 * 16 + lan16;
#pragma unroll
        for (int cpl = 0; cpl < KSTEPS; ++cpl) {
            const v2f b = Bp[cpl * RC];                 // ds_load_b64, imm offset
#pragma unroll
            for (int t = 0; t < TPW; ++t) {
                const v2f a = Ap[cpl * COUT + t * 64];  // fuses to ds_load_2addr_stride64_b64
                acc[t] = __builtin_amdgcn_wmma_f32_16x16x4_f32(
                    false, a, false, b, (short)0, acc[t], false, false);
            }
        }
    };

#if USE_ASYNC_LDS
    issueW(0);          // kick off the async pipeline before input staging so the
    issueW(1);          // weight DMA overlaps the patch load below
#endif

    // ---- stage input patch X[n][ci][h-1..h+1][wB-1..wB+32] -> LDS (pair-interleaved)
    {
        const float* xn = x + (size_t)n * CIN * HH * WWID;
        for (int i = tid; i < CIN * RC; i += 256) {
            const int ci = i / RC;
            const int rc = i % RC;
            const int r  = rc / PATCH;
            const int c  = rc % PATCH;
            const int gr = h + r - 1;
            const int gc = wB + c - 1;
            float v = 0.0f;
            if ((unsigned)gr < (unsigned)HH && (unsigned)gc < (unsigned)WWID)
                v = xn[(size_t)ci * (HH * WWID) + gr * WWID + gc];
            sX[((ci >> 1) * RC + rc) * 2 + (ci & 1)] = v;
        }
    }

    // ---- pipelined phase loop: 2 phases in flight on ASYNCcnt
    for (int p = 0; p < NPH - 1; ++p) {
#if USE_ASYNC_LDS
        __builtin_amdgcn_s_wait_asynccnt(PHOPS);  // oldest phase (p) retired, in-order
        __syncthreads();                          // its LDS writes visible to all waves
        compute(p);
        __syncthreads();                          // all waves done reading buf[p&1]
        if (p + 2 < NPH) issueW(p + 2);           // refill the buffer just vacated
#else
        __syncthreads();
        copyW(p);
        __syncthreads();
        compute(p);
#endif
    }
    // final phase: must wait the counter all the way to 0
#if USE_ASYNC_LDS
    __builtin_amdgcn_s_wait_asynccnt(0);
    __syncthreads();
    compute(NPH - 1);
#else
    __syncthreads();
    copyW(NPH - 1);
    __syncthreads();
    compute(NPH - 1);
#endif

    // ---- epilogue: D VGPR j: lanes0-15 -> M=j (co), N=lane (pix); lanes16-31 -> M=j+8
    const int coOff = hi16 * 8;
#pragma unroll
    for (int t = 0; t < TPW; ++t) {
        const int coBase = (coT0 + t * 4) * 16;
#pragma unroll
        for (int j = 0; j < 8; ++j) {
            const int co = coBase + coOff + j;
            out[(((size_t)n * COUT + co) * HH + h) * WWID + wB + wSub * 16 + lan16]
                = acc[t][j];
        }
    }
}

extern "C" void kernel_launch(void* const* d_in, const int* in_sizes, int n_in,
                              void* d_out, int out_size, void* d_ws, size_t ws_size,
                              hipStream_t stream) {
    const float* x  = (const float*)d_in[0];
    const float* w1 = (const float*)d_in[1];
    const float* w2 = (const float*)d_in[2];
    const float* w3 = (const float*)d_in[3];
    float* outp = (float*)d_out;

    // Workspace: two fp32 intermediates (16,128,224,224) + transformed weights.
    const size_t midElems = (size_t)NB * 128 * HH * WWID;
    float* t1  = (float*)d_ws;
    float* t2  = t1 + midElems;
    float* wt1 = t2 + midElems;                   // 128*64*9  = 73728 floats
    float* wt2 = wt1 + (size_t)128 * 64 * 9;      // 128*128*9 = 147456 floats
    float* wt3 = wt2 + (size_t)128 * 128 * 9;     // 64*128*9  = 73728 floats

    wtransform<64, 128><<<dim3((128 * 64 * 9  + 255) / 256), dim3(256), 0, stream>>>(w1, wt1);
    wtransform<128,128><<<dim3((128 * 128 * 9 + 255) / 256), dim3(256), 0, stream>>>(w2, wt2);
    wtransform<128, 64><<<dim3((64 * 128 * 9  + 255) / 256), dim3(256), 0, stream>>>(w3, wt3);

    dim3 grid(WWID / WTILE, HH, NB);   // (7, 224, 16)
    dim3 blk(256);
    conv3x3_wmma<64, 128><<<grid, blk, 0, stream>>>(x,  wt1, t1);
    conv3x3_wmma<128,128><<<grid, blk, 0, stream>>>(t1, wt2, t2);
    conv3x3_wmma<128, 64><<<grid, blk, 0, stream>>>(t2, wt3, outp);
}